// DistanceBasedQueryScorer_37520834298055
// MI455X (gfx1250) — compile-verified
//
#include <hip/hip_runtime.h>
#include <math.h>

// Problem constants (match the reference).
#define NQ 16384
#define NB 128
#define NF 64
#define HD 128
#define EPSF 1e-8f

typedef __attribute__((ext_vector_type(2))) float v2f;
typedef __attribute__((ext_vector_type(8))) float v8f;

// One block = 256 threads = 8 wave32 waves, handles a 16-query x 128-bin output tile.
__global__ __launch_bounds__(256) void scorer_kernel(
    const float* __restrict__ Q,        // [NQ][HD]
    const float* __restrict__ probes,   // [NB][HD]  (real = [:,:64], imag = [:,64:])
    const float* __restrict__ w_raw,    // [NB][NF]
    const float* __restrict__ w_mag,    // [NB][NF]
    const float* __restrict__ bias,     // [NB]
    float* __restrict__ out)            // [NQ][NB]
{
  __shared__ float sQr[16][NF];        // normalized Q real, 4 KB
  __shared__ float sQi[16][NF];        // normalized Q imag, 4 KB
  __shared__ float sQm[16][NF];        // Q magnitude, 4 KB (WMMA A source)
  __shared__ float sW[NB][NF + 4];     // -softplus(w_raw), padded vs bank conflicts, 34 KB
  __shared__ float sOut[16][NB];       // distance+bias partials, 8 KB

  const int tid = threadIdx.x;
  const int q0  = blockIdx.x * 16;
  const int row = tid >> 4;            // query within tile, 0..15
  const int col = tid & 15;            // 0..15

  // Warm the L2-resident probe table (emits global_prefetch_b8).
  __builtin_prefetch(&probes[col * HD], 0, 1);

  // ---- Phase 0: eff_w = -softplus(w_raw), staged in LDS ----
  for (int i = tid; i < NB * NF; i += 256) {
    const int b = i >> 6;
    const int f = i & 63;
    const float x  = w_raw[i];
    const float sp = fmaxf(x, 0.0f) + log1pf(expf(-fabsf(x)));  // stable softplus
    sW[b][f] = -sp;
  }

  // ---- Phase 1: load 16x128 Q tile, row-normalize, build Qmag ----
  float v[8];
  float ss = 0.0f;
  const float* qrow = Q + (size_t)(q0 + row) * HD;
#pragma unroll
  for (int k = 0; k < 8; ++k) {
    const float x = qrow[col + 16 * k];
    v[k] = x;
    ss = fmaf(x, x, ss);
  }
  // Reduce sum of squares across the 16 lanes sharing a row (wave32 halves).
#pragma unroll
  for (int off = 8; off >= 1; off >>= 1)
    ss += __shfl_xor(ss, off, 16);
  const float inv = 1.0f / (sqrtf(ss) + EPSF);
#pragma unroll
  for (int k = 0; k < 8; ++k) {
    const int d = col + 16 * k;
    const float xn = v[k] * inv;
    if (d < NF) sQr[row][d] = xn;
    else        sQi[row][d - NF] = xn;
  }
  // Qmag from this thread's own register values (v[k] -> real f, v[k+4] -> imag f).
#pragma unroll
  for (int k = 0; k < 4; ++k) {
    const float qr = v[k] * inv;
    const float qi = v[k + 4] * inv;
    sQm[row][col + 16 * k] = sqrtf(fmaf(qr, qr, fmaf(qi, qi, EPSF)));
  }
  __syncthreads();

  // ---- Phase 2: distance term (VALU/TRANS-bound core) ----
  float acc[8];
#pragma unroll
  for (int j = 0; j < 8; ++j) acc[j] = bias[col + 16 * j];

  for (int f = 0; f < NF; f += 4) {
    const float4 qr = *reinterpret_cast<const float4*>(&sQr[row][f]);
    const float4 qi = *reinterpret_cast<const float4*>(&sQi[row][f]);
#pragma unroll
    for (int j = 0; j < 8; ++j) {
      const int b = col + 16 * j;
      const float4 pr = *reinterpret_cast<const float4*>(&probes[b * HD + f]);
      const float4 pi = *reinterpret_cast<const float4*>(&probes[b * HD + NF + f]);
      const float4 ww = *reinterpret_cast<const float4*>(&sW[b][f]);
      { float er = pr.x - qr.x, ei = pi.x - qi.x;
        float s  = fmaf(er, er, fmaf(ei, ei, EPSF));
        acc[j] = fmaf(sqrtf(s), ww.x, acc[j]); }
      { float er = pr.y - qr.y, ei = pi.y - qi.y;
        float s  = fmaf(er, er, fmaf(ei, ei, EPSF));
        acc[j] = fmaf(sqrtf(s), ww.y, acc[j]); }
      { float er = pr.z - qr.z, ei = pi.z - qi.z;
        float s  = fmaf(er, er, fmaf(ei, ei, EPSF));
        acc[j] = fmaf(sqrtf(s), ww.z, acc[j]); }
      { float er = pr.w - qr.w, ei = pi.w - qi.w;
        float s  = fmaf(er, er, fmaf(ei, ei, EPSF));
        acc[j] = fmaf(sqrtf(s), ww.w, acc[j]); }
    }
  }
#pragma unroll
  for (int j = 0; j < 8; ++j) sOut[row][col + 16 * j] = acc[j];
  __syncthreads();

  // ---- Phase 3: magnitude GEMM via V_WMMA_F32_16X16X4_F32 ----
  // Wave wv owns bins [16*wv, 16*wv+16). C/D layout: lane -> N = lane&15,
  // VGPR r -> M = r + 8*(lane>>4). A layout: lane -> M = lane&15,
  // VGPR {0,1} hold K = {0,1} (lanes 0-15) / K = {2,3} (lanes 16-31).
  const int lane = tid & 31;
  const int wv   = tid >> 5;           // 0..7
  const int n16  = lane & 15;
  const int hi   = lane >> 4;          // 0 or 1
  const int bcol = wv * 16 + n16;      // bin column this lane produces

  v8f c = {0.f, 0.f, 0.f, 0.f, 0.f, 0.f, 0.f, 0.f};
#pragma unroll
  for (int k0 = 0; k0 < NF; k0 += 4) {
    const int ka = k0 + 2 * hi;        // this lane's K pair
    const float2 av = *reinterpret_cast<const float2*>(&sQm[n16][ka]);
    // B[k][n] = Wm^T[k][n] = w_mag[n][k] read with transposed indexing.
    const float2 bv = *reinterpret_cast<const float2*>(&w_mag[bcol * NF + ka]);
    v2f a;  a[0]  = av.x;  a[1]  = av.y;
    v2f bb; bb[0] = bv.x;  bb[1] = bv.y;
    c = __builtin_amdgcn_wmma_f32_16x16x4_f32(
        /*neg_a=*/false, a, /*neg_b=*/false, bb,
        /*c_mod=*/(short)0, c, /*reuse_a=*/false, /*reuse_b=*/false);
  }

#pragma unroll
  for (int r = 0; r < 8; ++r) {
    const int m = r + 8 * hi;
    out[(size_t)(q0 + m) * NB + bcol] = sOut[m][bcol] + c[r];
  }
}

extern "C" void kernel_launch(void* const* d_in, const int* in_sizes, int n_in,
                              void* d_out, int out_size, void* d_ws, size_t ws_size,
                              hipStream_t stream) {
  (void)in_sizes; (void)n_in; (void)d_ws; (void)ws_size; (void)out_size;
  const float* Q      = (const float*)d_in[0];
  const float* probes = (const float*)d_in[1];
  const float* w_raw  = (const float*)d_in[2];
  const float* w_mag  = (const float*)d_in[3];
  const float* bias   = (const float*)d_in[4];
  float* out = (float*)d_out;

  scorer_kernel<<<NQ / 16, 256, 0, stream>>>(Q, probes, w_raw, w_mag, bias, out);
}